// Colorizer_51316269253015
// MI455X (gfx1250) — compile-verified
//
#include <hip/hip_runtime.h>
#include <math.h>

typedef __attribute__((ext_vector_type(2))) float v2f;
typedef __attribute__((ext_vector_type(8))) float v8f;

constexpr int H = 128, W = 128, C = 128, HW = H * W;
constexpr int NREF = 3;
constexpr int RR = 6;          // patch radius
constexpr int PS = 13;         // patch size (2R+1)
constexpr int LSTRIDE = 33;    // LDS row stride (pad to avoid bank conflicts)

// -------------------------------------------------------------------------
// Kernel 1: fused 13x13 correlation (f32 WMMA) + argmax over the 169 taps.
// One wave = 16 target pixels of one row for one ref. Block = 4 waves.
// Grid = (W/64, H, NREF).
// All loads in the hot loop are unconditional (clamped addresses); zero
// padding is applied as a column mask at tap-extraction time, which is
// mathematically identical to zeroing the B columns before the GEMM.
// -------------------------------------------------------------------------
__global__ __launch_bounds__(128)
void corr_argmax_kernel(const float* __restrict__ feats_r,
                        const float* __restrict__ feats_t,
                        float* __restrict__ qval,
                        int*   __restrict__ qidx)
{
    __shared__ float lds[4][16 * LSTRIDE];

    const int wave = threadIdx.x >> 5;
    const int lane = threadIdx.x & 31;
    const int half = lane >> 4;     // lane half selects K offset (+2) per ISA layout
    const int n16  = lane & 15;     // M (target pixel) / N (ref position) index

    const int r  = blockIdx.z;
    const int y  = blockIdx.y;
    const int x0 = blockIdx.x * 64 + wave * 16;

    const float* __restrict__ fr = feats_r + (size_t)r * C * HW;

    // ---- Preload A fragments: A[i,k] = feats_t[k, y, x0+i] ----
    // f32 16x4 A layout: lanes 0-15 -> K = {0,1}, lanes 16-31 -> K = {2,3}
    v2f a[32];
    {
        const float* base = feats_t + (size_t)y * W + (x0 + n16);
        #pragma unroll
        for (int kk = 0; kk < 32; ++kk) {
            const int ch = 4 * kk + 2 * half;
            a[kk].x = base[(size_t)ch * HW];
            a[kk].y = base[(size_t)(ch + 1) * HW];
        }
    }

    // Ref-position x for B tile 0 (j = n16) and tile 1 (j = n16+16),
    // clamped in-range so loads are always legal and unconditional.
    const int xb0 = x0 - RR + n16;              // in [-6, 121]
    const int xc0 = (xb0 < 0) ? 0 : xb0;
    const int xb1 = xb0 + 16;                   // in [10, 137]
    const int xc1 = (xb1 > W - 1) ? (W - 1) : xb1;

    // Per-lane tap mask: tap dxi of pixel (x0+n16) reads ref x = xb0+dxi.
    float xmask[PS];
    #pragma unroll
    for (int dxi = 0; dxi < PS; ++dxi) {
        const int xr = xb0 + dxi;
        xmask[dxi] = (xr >= 0 && xr < W) ? 1.0f : 0.0f;
    }

    float  best  = -3.4e38f;
    int    bestn = 0;
    float* mt    = lds[wave];

    #pragma unroll 1
    for (int dy = 0; dy < PS; ++dy) {
        const int  yr    = y + dy - RR;
        const bool rowok = (yr >= 0) && (yr < H);   // uniform across block

        if (rowok) {
            const float* p0 = fr + (size_t)yr * W;

            // Hint the next row into cache (global_prefetch_b8).
            __builtin_prefetch(p0 + W + xc0, 0, 3);

            v8f acc0 = {};
            v8f acc1 = {};
            #pragma unroll
            for (int kk = 0; kk < 32; ++kk) {
                const size_t choff = (size_t)(4 * kk + 2 * half) * HW;
                v2f b0, b1;
                b0.x = p0[choff + xc0];         // channel ch,   x = xc0
                b0.y = p0[choff + HW + xc0];    // channel ch+1, x = xc0
                b1.x = p0[choff + xc1];
                b1.y = p0[choff + HW + xc1];
                acc0 = __builtin_amdgcn_wmma_f32_16x16x4_f32(
                           false, a[kk], false, b0, (short)0, acc0, false, false);
                acc1 = __builtin_amdgcn_wmma_f32_16x16x4_f32(
                           false, a[kk], false, b1, (short)0, acc1, false, false);
            }

            // Spill M (16x32) to LDS. C/D layout: VGPR v -> row, lane half -> +8.
            #pragma unroll
            for (int v = 0; v < 8; ++v) {
                const int row = v + 8 * half;
                mt[row * LSTRIDE + n16]      = acc0[v];
                mt[row * LSTRIDE + 16 + n16] = acc1[v];
            }

            // Per-pixel argmax update; both halves redundantly own pixel n16
            // (keeps EXEC full, results identical). Mask applies zero padding.
            #pragma unroll
            for (int dxi = 0; dxi < PS; ++dxi) {
                const float val = mt[n16 * LSTRIDE + n16 + dxi] * xmask[dxi];
                const int   n   = dy * PS + dxi;
                if (val > best) { best = val; bestn = n; }  // strict > = first idx
            }
        } else {
            // Entire row is zero padding: 13 taps of exactly 0.0. Only the
            // first can win (0 > 0 is false afterwards) -> first-index kept.
            if (0.0f > best) { best = 0.0f; bestn = dy * PS; }
        }
    }

    if (half == 0) {
        const int p = y * W + x0 + n16;
        qval[(size_t)r * HW + p] = best;
        qidx[(size_t)r * HW + p] = bestn;
    }
}

// -------------------------------------------------------------------------
// Kernel 2: gather argmax patch element from ::4-subsampled quantized_r,
// softmax over refs, weighted blend. One thread per pixel.
// Loads are unconditional (clamped) with a mask multiply; values in [0,1).
// -------------------------------------------------------------------------
__global__ __launch_bounds__(256)
void finalize_kernel(const float* __restrict__ quantized_r,
                     const float* __restrict__ qval,
                     const int*   __restrict__ qidx,
                     float* __restrict__ out)
{
    const int p = blockIdx.x * blockDim.x + threadIdx.x;
    if (p >= HW) return;
    const int y = p / W;
    const int x = p % W;

    float v[NREF];
    float g[NREF][3];
    #pragma unroll
    for (int r = 0; r < NREF; ++r) {
        v[r] = qval[(size_t)r * HW + p];
        const int n  = qidx[(size_t)r * HW + p];
        int ry = y + n / PS - RR;
        int rx = x + n % PS - RR;
        const float okf = (ry >= 0 && ry < H && rx >= 0 && rx < W) ? 1.0f : 0.0f;
        ry = (ry < 0) ? 0 : ((ry > H - 1) ? H - 1 : ry);
        rx = (rx < 0) ? 0 : ((rx > W - 1) ? W - 1 : rx);
        #pragma unroll
        for (int c = 0; c < 3; ++c) {
            // quantized_r[r, 0, c, ry*4, rx*4] with shape [3,1,3,512,512]
            const size_t idx = (((size_t)(r * 3 + c) * (H * 4) + (size_t)ry * 4)
                                * (W * 4)) + (size_t)rx * 4;
            g[r][c] = quantized_r[idx] * okf;
        }
    }

    const float m  = fmaxf(v[0], fmaxf(v[1], v[2]));
    const float e0 = expf(v[0] - m);
    const float e1 = expf(v[1] - m);
    const float e2 = expf(v[2] - m);
    const float inv = 1.0f / (e0 + e1 + e2);

    #pragma unroll
    for (int c = 0; c < 3; ++c) {
        out[(size_t)c * HW + p] =
            (e0 * g[0][c] + e1 * g[1][c] + e2 * g[2][c]) * inv;
    }
}

// -------------------------------------------------------------------------
// Launch
// -------------------------------------------------------------------------
extern "C" void kernel_launch(void* const* d_in, const int* in_sizes, int n_in,
                              void* d_out, int out_size, void* d_ws, size_t ws_size,
                              hipStream_t stream)
{
    const float* feats_r     = (const float*)d_in[0]; // [3,1,128,128,128]
    const float* feats_t     = (const float*)d_in[1]; // [1,128,128,128]
    const float* quantized_r = (const float*)d_in[2]; // [3,1,3,512,512]
    (void)in_sizes; (void)n_in; (void)out_size; (void)ws_size;

    float* qval = (float*)d_ws;                                   // 3*16384 f32
    int*   qidx = (int*)((char*)d_ws + (size_t)NREF * HW * sizeof(float));
    float* out  = (float*)d_out;                                  // [1,3,128,128]

    dim3 grid1(W / 64, H, NREF);   // (2, 128, 3), 4 waves/block
    corr_argmax_kernel<<<grid1, 128, 0, stream>>>(feats_r, feats_t, qval, qidx);

    finalize_kernel<<<(HW + 255) / 256, 256, 0, stream>>>(quantized_r, qval, qidx, out);
}